// Model_7765300871331
// MI455X (gfx1250) — compile-verified
//
#include <hip/hip_runtime.h>

#define N_NODES 65536
#define N_EDGES 1048576
#define F 128

typedef __attribute__((ext_vector_type(2))) float v2f;
typedef __attribute__((ext_vector_type(8))) float v8f;

// ---------------- utility kernels ----------------

__global__ __launch_bounds__(256) void k_zero(float* __restrict__ p, int n) {
    int i = blockIdx.x * 256 + threadIdx.x;
    if (i < n) p[i] = 0.0f;
}

__global__ __launch_bounds__(256) void k_degrees(const int* __restrict__ src,
                                                 const int* __restrict__ dst,
                                                 float* __restrict__ deg_out,
                                                 float* __restrict__ deg_in) {
    int e = blockIdx.x * 256 + threadIdx.x;
    if (e < N_EDGES) {
        atomicAdd(&deg_out[src[e]], 1.0f);
        atomicAdd(&deg_in[dst[e]], 1.0f);
    }
}

// x[i] = rsqrt(max(x[i], 1))  -- applied to both degree arrays at once
__global__ __launch_bounds__(256) void k_rsqrt_clamp(float* __restrict__ x, int n) {
    int i = blockIdx.x * 256 + threadIdx.x;
    if (i < n) x[i] = rsqrtf(fmaxf(x[i], 1.0f));
}

// out[i] = (relu?) (agg[i]*in_norm[node] + b[f])
__global__ __launch_bounds__(256) void k_pointwise(const float* __restrict__ agg,
                                                   const float* __restrict__ in_norm,
                                                   const float* __restrict__ bias,
                                                   float* __restrict__ out,
                                                   int do_relu) {
    int i = blockIdx.x * 256 + threadIdx.x;   // over N_NODES*F = 8388608
    int node = i >> 7;
    int f = i & (F - 1);
    float v = agg[i] * in_norm[node] + bias[f];
    if (do_relu) v = fmaxf(v, 0.0f);
    out[i] = v;
}

// ---------------- edge scatter (segment_sum) ----------------
// one wave (32 lanes) per edge, float4 per lane -> 4 atomic f32 adds
__global__ __launch_bounds__(256) void k_scatter_add(const float* __restrict__ H,
                                                     float* __restrict__ acc,
                                                     const int* __restrict__ src,
                                                     const int* __restrict__ dst) {
    long long gid = (long long)blockIdx.x * 256 + threadIdx.x;
    int e = (int)(gid >> 5);
    int lane = (int)(gid & 31);
    int s = src[e];
    int d = dst[e];
    const float4 v = *(const float4*)(H + (size_t)s * F + lane * 4);
    float* p = acc + (size_t)d * F + lane * 4;
    atomicAdd(p + 0, v.x);
    atomicAdd(p + 1, v.y);
    atomicAdd(p + 2, v.z);
    atomicAdd(p + 3, v.w);
}

__global__ __launch_bounds__(256) void k_scatter_add_w(const float* __restrict__ H,
                                                       float* __restrict__ acc,
                                                       const int* __restrict__ src,
                                                       const int* __restrict__ dst,
                                                       const float* __restrict__ ew) {
    long long gid = (long long)blockIdx.x * 256 + threadIdx.x;
    int e = (int)(gid >> 5);
    int lane = (int)(gid & 31);
    int s = src[e];
    int d = dst[e];
    float w = ew[e];
    const float4 v = *(const float4*)(H + (size_t)s * F + lane * 4);
    float* p = acc + (size_t)d * F + lane * 4;
    atomicAdd(p + 0, v.x * w);
    atomicAdd(p + 1, v.y * w);
    atomicAdd(p + 2, v.z * w);
    atomicAdd(p + 3, v.w * w);
}

// ---------------- WMMA f32 GEMM: Y = (rowScale * X) @ W ----------------
// X: [N_NODES, F], W: [F, F] row-major, Y: [N_NODES, F]
// grid = (N_NODES/16/8, F/16), block = 256 (8 waves). Wave w computes the
// 16x16 C tile at rowTile = blockIdx.x*8 + w, colTile = blockIdx.y.
// K-loop: 32 steps of v_wmma_f32_16x16x4_f32.
__global__ __launch_bounds__(256) void k_gemm_wmma(const float* __restrict__ X,
                                                   const float* __restrict__ rowScale,
                                                   const float* __restrict__ W,
                                                   float* __restrict__ Y) {
    __shared__ float Wl[F * 16];  // W[:, colBase:colBase+16], layout [k][n0]

    const int tid = threadIdx.x;
    const int wave = tid >> 5;
    const int lane = tid & 31;
    const int l15 = lane & 15;
    const int colBase = blockIdx.y * 16;

    // stage the 16-column slice of W into LDS (8 KB)
    for (int i = tid; i < F * 16; i += 256) {
        int k = i >> 4;
        int n0 = i & 15;
        Wl[i] = W[k * F + colBase + n0];
    }
    __syncthreads();

    const int rowBase = (blockIdx.x * 8 + wave) * 16;
    const int mrow = rowBase + l15;          // A-matrix row for this lane
    const int khalf = (lane >> 4) * 2;       // 0 for lanes 0-15, 2 for 16-31
    const float scale = rowScale[mrow];
    const float* xrow = X + (size_t)mrow * F;

    v8f acc = {};
    #pragma unroll 4
    for (int k0 = 0; k0 < F; k0 += 4) {
        // A fragment (16x4 f32): VGPR0 = K=k0 / k0+2, VGPR1 = K=k0+1 / k0+3
        v2f a;
        a.x = xrow[k0 + khalf] * scale;
        a.y = xrow[k0 + khalf + 1] * scale;
        // B fragment (4x16 f32): row K striped across lanes within a VGPR
        v2f b;
        b.x = Wl[(k0 + khalf) * 16 + l15];
        b.y = Wl[(k0 + khalf + 1) * 16 + l15];
        acc = __builtin_amdgcn_wmma_f32_16x16x4_f32(
            /*neg_a=*/false, a, /*neg_b=*/false, b,
            /*c_mod=*/(short)0, acc, /*reuse_a=*/false, /*reuse_b=*/false);
    }

    // C layout: VGPR v -> M = v + (lane>=16 ? 8 : 0), N = lane&15
    const int row0 = rowBase + (lane >> 4) * 8;
    const int col = colBase + l15;
    #pragma unroll
    for (int v = 0; v < 8; ++v) {
        Y[(size_t)(row0 + v) * F + col] = acc[v];
    }
}

// ---------------- driver ----------------

extern "C" void kernel_launch(void* const* d_in, const int* in_sizes, int n_in,
                              void* d_out, int out_size, void* d_ws, size_t ws_size,
                              hipStream_t stream) {
    const float* feat = (const float*)d_in[0];
    const float* ew   = (const float*)d_in[1];
    const float* W1   = (const float*)d_in[2];
    const float* b1   = (const float*)d_in[3];
    const float* W2   = (const float*)d_in[4];
    const float* b2   = (const float*)d_in[5];
    const int*   src  = (const int*)d_in[6];
    const int*   dst  = (const int*)d_in[7];
    float* out = (float*)d_out;

    float* ws   = (float*)d_ws;
    float* degO = ws;                       // [N_NODES] -> out_norm
    float* degI = ws + N_NODES;             // [N_NODES] -> in_norm
    float* bufA = ws + 2 * N_NODES;         // [N_NODES*F]
    float* bufB = bufA + (size_t)N_NODES * F;

    const int NF = N_NODES * F;             // 8388608
    dim3 gemm_grid(N_NODES / 16 / 8, F / 16);
    const int scatter_blocks = (N_EDGES * 32) / 256;

    // degrees + norms
    k_zero<<<(2 * N_NODES) / 256, 256, 0, stream>>>(degO, 2 * N_NODES);
    k_degrees<<<N_EDGES / 256, 256, 0, stream>>>(src, dst, degO, degI);
    k_rsqrt_clamp<<<(2 * N_NODES) / 256, 256, 0, stream>>>(degO, 2 * N_NODES);

    // layer 1: bufA = (feat*out_norm)@W1 ; bufB = segsum ; bufA = relu(norm+b1)
    k_gemm_wmma<<<gemm_grid, 256, 0, stream>>>(feat, degO, W1, bufA);
    k_zero<<<NF / 256, 256, 0, stream>>>(bufB, NF);
    k_scatter_add<<<scatter_blocks, 256, 0, stream>>>(bufA, bufB, src, dst);
    k_pointwise<<<NF / 256, 256, 0, stream>>>(bufB, degI, b1, bufA, 1);

    // layer 2: bufB = (h1*out_norm)@W2 ; bufA = segsum ; bufB = norm+b2
    k_gemm_wmma<<<gemm_grid, 256, 0, stream>>>(bufA, degO, W2, bufB);
    k_zero<<<NF / 256, 256, 0, stream>>>(bufA, NF);
    k_scatter_add<<<scatter_blocks, 256, 0, stream>>>(bufB, bufA, src, dst);
    k_pointwise<<<NF / 256, 256, 0, stream>>>(bufA, degI, b2, bufB, 0);

    // final u_mul_e + segment_sum into d_out
    k_zero<<<NF / 256, 256, 0, stream>>>(out, NF);
    k_scatter_add_w<<<scatter_blocks, 256, 0, stream>>>(bufB, out, src, dst, ew);
}